// GraphVAE_41300405518366
// MI455X (gfx1250) — compile-verified
//
#include <hip/hip_runtime.h>
#include <hip/hip_bf16.h>
#include <math.h>

#define N_CELLS 10000
#define E_EDGES 320000
#define IN_DIM  2000
#define H_DIM   512
#define L_DIM   64
#define BN_EPS_F 1e-5f

typedef __attribute__((ext_vector_type(2))) float v2f;
typedef __attribute__((ext_vector_type(8))) float v8f;
typedef unsigned int v4u __attribute__((ext_vector_type(4)));
typedef int          v4i_t __attribute__((ext_vector_type(4)));
typedef int          v8i_t __attribute__((ext_vector_type(8)));

// Tensor Data Mover path. This toolchain exposes the 6-arg builtin
// (uint32x4, int32x8, int32x4, int32x4, int32x8, i32 cpol) — confirmed by the
// round-2 diagnostic ("expected 6, have 5"). Guarded so the host pass and
// TDM-less toolchains fall back to clamp+select vector loads.
#if __has_builtin(__builtin_amdgcn_tensor_load_to_lds) && \
    __has_builtin(__builtin_amdgcn_s_wait_tensorcnt)
#define USE_TDM 1
#else
#define USE_TDM 0
#endif

#define BM 64
#define BN 32
#define BK 32
#define LDP 36   /* padded LDS row: 32 data + 4 pad floats -> 144B, 16B aligned */

#if USE_TDM
__device__ __forceinline__ unsigned lds_addr_of(const void* p) {
  return (unsigned)(unsigned long long)(__attribute__((address_space(3))) const void*)p;
}

// 2D tile DMA: tile_cols (dim0) x tile_rows (dim1), fp32, row stride ld.
// rem_cols/rem_rows = tensor extent from tile start (hardware zero-fills OOB).
// LDS padding: after every 32 DWORDs insert 4 DWORDs (matches LDP=36 rows).
__device__ __forceinline__ void tdm_load_tile(const float* gptr, unsigned lds_byte,
                                              int tile_rows, int tile_cols, int ld,
                                              int rem_cols, int rem_rows) {
  unsigned long long ga = (unsigned long long)(const void*)gptr;
  v4u g0;
  g0.x = 1u;                                              // count=1, user D#
  g0.y = lds_byte;                                        // lds_addr
  g0.z = (unsigned)(ga & 0xFFFFFFFFu);                    // global_addr[31:0]
  g0.w = (unsigned)((ga >> 32) & 0x01FFFFFFu) | (2u << 30); // addr[56:32] | type=2
  v8i_t g1;
  // data_size=4B (2<<16) | pad_enable (1<<20) | pad_interval=32dw (4<<22) | pad=4dw (3<<25)
  g1[0] = (int)((2u << 16) | (1u << 20) | (4u << 22) | (3u << 25));
  g1[1] = (int)(((unsigned)rem_cols & 0xFFFFu) << 16);    // tensor_dim0[15:0]
  g1[2] = (int)((((unsigned)rem_cols >> 16) & 0xFFFFu) |
                (((unsigned)rem_rows & 0xFFFFu) << 16));  // dim0[31:16] | dim1[15:0]
  g1[3] = (int)((((unsigned)rem_rows >> 16) & 0xFFFFu) |
                (((unsigned)tile_cols) << 16));           // dim1[31:16] | tile_dim0
  g1[4] = (int)((unsigned)tile_rows & 0xFFFFu);           // tile_dim1, tile_dim2=0
  g1[5] = (int)(unsigned)ld;                              // tensor_dim0_stride[31:0]
  g1[6] = 0; g1[7] = 0;
  v4i_t g2 = {0, 0, 0, 0};   // group 2: unused (2D tile)
  v4i_t g3 = {0, 0, 0, 0};   // group 3: unused (2D tile)
  v8i_t g4 = {0, 0, 0, 0, 0, 0, 0, 0};
  __builtin_amdgcn_tensor_load_to_lds(g0, g1, g2, g3, g4, 0);
}
#endif

// ---------------------------------------------------------------------------
// WMMA fp32 GEMM: C[M,Nc] = epi(A[M,K] @ B[K,Nc] (+bias))
// 256 threads = 8 waves; tile BM=64 x BN=32; K-tile 32; one 16x16 subtile/wave
// via V_WMMA_F32_16X16X4_F32.  All K values are multiples of 4 and all row
// strides are 16B-aligned, so staging uses branch-free clamp+select b128 loads
// (or the TDM DMA path).  EPI: 0=none,1=bias,2=bias+relu,3=sigmoid.
// BT: B is [Nc,K] row-major (i.e. multiply by B^T).
// ---------------------------------------------------------------------------
template <int EPI, bool BT>
__global__ __launch_bounds__(256) void gemm_wmma_f32(
    const float* __restrict__ A, const float* __restrict__ B,
    const float* __restrict__ bias, float* __restrict__ C,
    int M, int Nc, int K)
{
  __shared__ float As[BM][LDP];
  __shared__ float Bs[BK][LDP];

  const int tid  = threadIdx.x;
  const int lane = tid & 31;
  const int wave = tid >> 5;            // 0..7
  const int wm   = (wave & 3) * 16;     // 0,16,32,48
  const int wn   = (wave >> 2) * 16;    // 0,16

  const int m0 = blockIdx.x * BM;
  const int n0 = blockIdx.y * BN;

  v8f acc = {};
  const int row  = lane & 15;
  const int koff = (lane >> 4) * 2;

  for (int k0 = 0; k0 < K; k0 += BK) {
#if USE_TDM
    if (wave == 0) {
      tdm_load_tile(A + (long long)m0 * K + k0, lds_addr_of(&As[0][0]),
                    BM, BK, K, K - k0, M - m0);
      if (!BT)
        tdm_load_tile(B + (long long)k0 * Nc + n0, lds_addr_of(&Bs[0][0]),
                      BK, BN, Nc, Nc - n0, K - k0);
      __builtin_amdgcn_s_wait_tensorcnt(0);
    }
    if (BT) {
      // B^T staging: thread -> 4 consecutive k of one n-row (branch-free)
      int nl = tid >> 3, k4 = (tid & 7) << 2;
      long long gn = n0 + nl; int gk = k0 + k4;
      bool ok = (gn < Nc) && (gk < K);
      long long rr = (gn < Nc) ? gn : (long long)Nc - 1;
      int cc = (gk < K) ? gk : K - 4;
      float4 v = *(const float4*)(B + rr * K + cc);
      if (!ok) v = make_float4(0.f, 0.f, 0.f, 0.f);
      Bs[k4 + 0][nl] = v.x; Bs[k4 + 1][nl] = v.y;
      Bs[k4 + 2][nl] = v.z; Bs[k4 + 3][nl] = v.w;
    }
#else
    // stage A tile (64x32): 512 float4 / 256 threads, clamp+select (no branches)
    #pragma unroll
    for (int i = 0; i < 2; ++i) {
      int f = tid + i * 256;
      int r = f >> 3, c4 = (f & 7) << 2;
      long long gr = m0 + r; int gc = k0 + c4;
      bool ok = (gr < M) && (gc < K);
      long long rr = (gr < M) ? gr : (long long)M - 1;
      int cc = (gc < K) ? gc : K - 4;
      float4 v = *(const float4*)(A + rr * K + cc);
      if (!ok) v = make_float4(0.f, 0.f, 0.f, 0.f);
      *(float4*)&As[r][c4] = v;
    }
    if (!BT) {
      int r = tid >> 3, c4 = (tid & 7) << 2;
      int gk = k0 + r, gn = n0 + c4;
      bool ok = (gk < K) && (gn < Nc);
      long long rr = (gk < K) ? gk : K - 1;
      int cc = (gn < Nc) ? gn : Nc - 4;
      float4 v = *(const float4*)(B + rr * Nc + cc);
      if (!ok) v = make_float4(0.f, 0.f, 0.f, 0.f);
      *(float4*)&Bs[r][c4] = v;
    } else {
      int nl = tid >> 3, k4 = (tid & 7) << 2;
      long long gn = n0 + nl; int gk = k0 + k4;
      bool ok = (gn < Nc) && (gk < K);
      long long rr = (gn < Nc) ? gn : (long long)Nc - 1;
      int cc = (gk < K) ? gk : K - 4;
      float4 v = *(const float4*)(B + rr * K + cc);
      if (!ok) v = make_float4(0.f, 0.f, 0.f, 0.f);
      Bs[k4 + 0][nl] = v.x; Bs[k4 + 1][nl] = v.y;
      Bs[k4 + 2][nl] = v.z; Bs[k4 + 3][nl] = v.w;
    }
#endif
    // prefetch next A tile rows into cache (global_prefetch_b8)
    if (k0 + BK < K) {
      long long pr = m0 + (tid & 63); if (pr >= M) pr = M - 1;
      __builtin_prefetch(A + pr * K + (k0 + BK), 0, 1);
    }
    __syncthreads();

    #pragma unroll
    for (int kt = 0; kt < BK; kt += 4) {
      v2f a, b;
      a.x = As[wm + row][kt + koff];
      a.y = As[wm + row][kt + koff + 1];
      b.x = Bs[kt + koff][wn + row];
      b.y = Bs[kt + koff + 1][wn + row];
      acc = __builtin_amdgcn_wmma_f32_16x16x4_f32(
          false, a, false, b, (short)0, acc, false, false);
    }
    __syncthreads();
  }

  const int col   = n0 + wn + (lane & 15);
  const int rbase = m0 + wm + ((lane >> 4) << 3);
  float bv = 0.0f;
  if ((EPI == 1 || EPI == 2) && col < Nc) bv = bias[col];

  if (rbase + 8 <= M && col < Nc) {           // fast interior path
    #pragma unroll
    for (int i = 0; i < 8; ++i) {
      float v = acc[i];
      if (EPI == 1) v += bv;
      if (EPI == 2) v = fmaxf(v + bv, 0.0f);
      if (EPI == 3) v = 1.0f / (1.0f + __expf(-v));
      C[(long long)(rbase + i) * Nc + col] = v;
    }
  } else {
    #pragma unroll
    for (int i = 0; i < 8; ++i) {
      int r = rbase + i;
      if (r < M && col < Nc) {
        float v = acc[i];
        if (EPI == 1) v += bv;
        if (EPI == 2) v = fmaxf(v + bv, 0.0f);
        if (EPI == 3) v = 1.0f / (1.0f + __expf(-v));
        C[(long long)r * Nc + col] = v;
      }
    }
  }
}

// ---------------------------------------------------------------------------
// Graph / elementwise kernels
// ---------------------------------------------------------------------------
__global__ void fill_f32(float* p, float v, long long n) {
  long long i = (long long)blockIdx.x * blockDim.x + threadIdx.x;
  if (i < n) p[i] = v;
}

__global__ void deg_count(const long long* __restrict__ dst, float* deg) {
  long long e = (long long)blockIdx.x * blockDim.x + threadIdx.x;
  if (e < E_EDGES) atomicAdd(&deg[(int)dst[e]], 1.0f);
}

__global__ void finalize_dis(float* d) {
  int i = blockIdx.x * blockDim.x + threadIdx.x;
  if (i < N_CELLS) d[i] = rsqrtf(d[i]);
}

__global__ void gcn_self_init(const float* __restrict__ hW,
                              const float* __restrict__ dis,
                              const float* __restrict__ bias,
                              float* __restrict__ out, int F) {
  long long idx = (long long)blockIdx.x * blockDim.x + threadIdx.x;
  if (idx >= (long long)N_CELLS * F) return;
  int i = (int)(idx / F), j = (int)(idx % F);
  float d = dis[i];
  out[idx] = hW[idx] * d * d + bias[j];
}

__global__ void gcn_edge_scatter(const long long* __restrict__ src,
                                 const long long* __restrict__ dst,
                                 const float* __restrict__ dis,
                                 const float* __restrict__ hW,
                                 float* __restrict__ out, int F) {
  long long idx = (long long)blockIdx.x * blockDim.x + threadIdx.x;
  int per_edge = F >> 2;
  long long e = idx / per_edge;
  int j = (int)(idx % per_edge) << 2;
  if (e >= E_EDGES) return;
  int s = (int)src[e], d = (int)dst[e];
  float w = dis[s] * dis[d];
  const float4 hv = *(const float4*)(hW + (long long)s * F + j);
  float* o = out + (long long)d * F + j;
  atomicAdd(o + 0, hv.x * w);
  atomicAdd(o + 1, hv.y * w);
  atomicAdd(o + 2, hv.z * w);
  atomicAdd(o + 3, hv.w * w);
}

__global__ __launch_bounds__(256) void bn_stats(const float* __restrict__ h,
                                                float* mean, float* var, int F) {
  int j = blockIdx.x;
  float s = 0.0f, s2 = 0.0f;
  for (int i = threadIdx.x; i < N_CELLS; i += 256) {
    float v = h[(long long)i * F + j];
    s += v; s2 += v * v;
  }
  __shared__ float sh[256], sh2[256];
  sh[threadIdx.x] = s; sh2[threadIdx.x] = s2; __syncthreads();
  for (int o = 128; o > 0; o >>= 1) {
    if (threadIdx.x < o) { sh[threadIdx.x] += sh[threadIdx.x + o];
                           sh2[threadIdx.x] += sh2[threadIdx.x + o]; }
    __syncthreads();
  }
  if (threadIdx.x == 0) {
    float m = sh[0] / (float)N_CELLS;
    mean[j] = m;
    var[j]  = sh2[0] / (float)N_CELLS - m * m;
  }
}

__global__ void bn_relu_apply(float* h, const float* __restrict__ mean,
                              const float* __restrict__ var,
                              const float* __restrict__ g,
                              const float* __restrict__ be, int F) {
  long long idx = (long long)blockIdx.x * blockDim.x + threadIdx.x;
  if (idx >= (long long)N_CELLS * F) return;
  int j = (int)(idx % F);
  float v = (h[idx] - mean[j]) * rsqrtf(var[j] + BN_EPS_F) * g[j] + be[j];
  h[idx] = fmaxf(v, 0.0f);
}

__global__ void reparam(const float* __restrict__ mu, const float* __restrict__ lv,
                        const float* __restrict__ eps, float* __restrict__ z) {
  long long idx = (long long)blockIdx.x * blockDim.x + threadIdx.x;
  if (idx < (long long)N_CELLS * L_DIM)
    z[idx] = mu[idx] + eps[idx] * __expf(0.5f * lv[idx]);
}

__global__ __launch_bounds__(256) void softmax_rows(float* x, int F) {
  int rrow = blockIdx.x;
  float* p = x + (long long)rrow * F;
  __shared__ float red[256];
  float m = -INFINITY;
  for (int j = threadIdx.x; j < F; j += 256) m = fmaxf(m, p[j]);
  red[threadIdx.x] = m; __syncthreads();
  for (int o = 128; o > 0; o >>= 1) {
    if (threadIdx.x < o) red[threadIdx.x] = fmaxf(red[threadIdx.x], red[threadIdx.x + o]);
    __syncthreads();
  }
  m = red[0]; __syncthreads();
  float s = 0.0f;
  for (int j = threadIdx.x; j < F; j += 256) {
    float e = __expf(p[j] - m);
    p[j] = e; s += e;
  }
  red[threadIdx.x] = s; __syncthreads();
  for (int o = 128; o > 0; o >>= 1) {
    if (threadIdx.x < o) red[threadIdx.x] += red[threadIdx.x + o];
    __syncthreads();
  }
  float inv = 1.0f / red[0];
  for (int j = threadIdx.x; j < F; j += 256) p[j] *= inv;
}

// ---------------------------------------------------------------------------
extern "C" void kernel_launch(void* const* d_in, const int* in_sizes, int n_in,
                              void* d_out, int out_size, void* d_ws, size_t ws_size,
                              hipStream_t stream) {
  (void)in_sizes; (void)n_in; (void)out_size; (void)ws_size;
  const int Nn = N_CELLS, H = H_DIM, L = L_DIM, IN = IN_DIM;

  const float*     x   = (const float*)d_in[0];
  const long long* ei  = (const long long*)d_in[1];   // [2,E] int64
  const float*     eps = (const float*)d_in[2];
  const float *W1=(const float*)d_in[3],  *b1=(const float*)d_in[4];
  const float *g1=(const float*)d_in[5],  *be1=(const float*)d_in[6];
  const float *W2=(const float*)d_in[7],  *b2=(const float*)d_in[8];
  const float *g2=(const float*)d_in[9],  *be2=(const float*)d_in[10];
  const float *Wmu=(const float*)d_in[11],*bmu=(const float*)d_in[12];
  const float *Wlv=(const float*)d_in[13],*blv=(const float*)d_in[14];
  const float *Wbil=(const float*)d_in[15];
  const float *Wd0=(const float*)d_in[16],*bd0=(const float*)d_in[17];
  const float *Wd1=(const float*)d_in[18],*bd1=(const float*)d_in[19];
  const float *Wd2=(const float*)d_in[20],*bd2=(const float*)d_in[21];

  const long long* src = ei;             // edge_index[0]
  const long long* dst = ei + E_EDGES;   // edge_index[1]

  float* ws    = (float*)d_ws;
  float* dis   = ws;                           // N
  float* stats = dis + Nn;                     // 2*H
  float* bufA  = stats + 2 * H;                // N*H
  float* bufB  = bufA + (long long)Nn * H;     // N*H
  float* zbuf  = bufB + (long long)Nn * H;     // N*L
  float* tbuf  = zbuf + (long long)Nn * L;     // N*L

  float* adj  = (float*)d_out;
  float* feat = adj  + (long long)Nn * Nn;
  float* muO  = feat + (long long)Nn * IN;
  float* lvO  = muO  + (long long)Nn * L;

  auto cdiv = [](long long a, long long b) { return (unsigned)((a + b - 1) / b); };

  // --- degree / dis ---
  fill_f32<<<cdiv(Nn, 256), 256, 0, stream>>>(dis, 1.0f, Nn);
  deg_count<<<cdiv(E_EDGES, 256), 256, 0, stream>>>(dst, dis);
  finalize_dis<<<cdiv(Nn, 256), 256, 0, stream>>>(dis);

  const unsigned gM = cdiv(Nn, BM);

  // --- encoder layer 1 ---
  gemm_wmma_f32<0,false><<<dim3(gM, cdiv(H, BN)), 256, 0, stream>>>(x, W1, nullptr, bufA, Nn, H, IN);
  gcn_self_init<<<cdiv((long long)Nn*H, 256), 256, 0, stream>>>(bufA, dis, b1, bufB, H);
  gcn_edge_scatter<<<cdiv((long long)E_EDGES*(H/4), 256), 256, 0, stream>>>(src, dst, dis, bufA, bufB, H);
  bn_stats<<<H, 256, 0, stream>>>(bufB, stats, stats + H, H);
  bn_relu_apply<<<cdiv((long long)Nn*H, 256), 256, 0, stream>>>(bufB, stats, stats + H, g1, be1, H);

  // --- encoder layer 2 ---
  gemm_wmma_f32<0,false><<<dim3(gM, cdiv(H, BN)), 256, 0, stream>>>(bufB, W2, nullptr, bufA, Nn, H, H);
  gcn_self_init<<<cdiv((long long)Nn*H, 256), 256, 0, stream>>>(bufA, dis, b2, bufB, H);
  gcn_edge_scatter<<<cdiv((long long)E_EDGES*(H/4), 256), 256, 0, stream>>>(src, dst, dis, bufA, bufB, H);
  bn_stats<<<H, 256, 0, stream>>>(bufB, stats, stats + H, H);
  bn_relu_apply<<<cdiv((long long)Nn*H, 256), 256, 0, stream>>>(bufB, stats, stats + H, g2, be2, H);

  // --- mu / logvar heads ---
  gemm_wmma_f32<0,false><<<dim3(gM, cdiv(L, BN)), 256, 0, stream>>>(bufB, Wmu, nullptr, tbuf, Nn, L, H);
  gcn_self_init<<<cdiv((long long)Nn*L, 256), 256, 0, stream>>>(tbuf, dis, bmu, muO, L);
  gcn_edge_scatter<<<cdiv((long long)E_EDGES*(L/4), 256), 256, 0, stream>>>(src, dst, dis, tbuf, muO, L);

  gemm_wmma_f32<0,false><<<dim3(gM, cdiv(L, BN)), 256, 0, stream>>>(bufB, Wlv, nullptr, tbuf, Nn, L, H);
  gcn_self_init<<<cdiv((long long)Nn*L, 256), 256, 0, stream>>>(tbuf, dis, blv, lvO, L);
  gcn_edge_scatter<<<cdiv((long long)E_EDGES*(L/4), 256), 256, 0, stream>>>(src, dst, dis, tbuf, lvO, L);

  // --- reparameterization ---
  reparam<<<cdiv((long long)Nn*L, 256), 256, 0, stream>>>(muO, lvO, eps, zbuf);

  // --- bilinear structure decoder: adj = sigmoid((z Wbil) z^T) ---
  gemm_wmma_f32<0,false><<<dim3(gM, cdiv(L, BN)), 256, 0, stream>>>(zbuf, Wbil, nullptr, tbuf, Nn, L, L);
  gemm_wmma_f32<3,true ><<<dim3(gM, cdiv(Nn, BN)), 256, 0, stream>>>(tbuf, zbuf, nullptr, adj, Nn, Nn, L);

  // --- feature decoder MLP + softmax ---
  gemm_wmma_f32<2,false><<<dim3(gM, cdiv(H, BN)), 256, 0, stream>>>(zbuf, Wd0, bd0, bufA, Nn, H, L);
  gemm_wmma_f32<2,false><<<dim3(gM, cdiv(H, BN)), 256, 0, stream>>>(bufA, Wd1, bd1, bufB, Nn, H, H);
  gemm_wmma_f32<1,false><<<dim3(gM, cdiv(IN, BN)), 256, 0, stream>>>(bufB, Wd2, bd2, feat, Nn, IN, H);
  softmax_rows<<<Nn, 256, 0, stream>>>(feat, IN);
}